// IIA_3272765079973
// MI455X (gfx1250) — compile-verified
//
#include <hip/hip_runtime.h>

typedef __attribute__((ext_vector_type(2))) float v2f;
typedef __attribute__((ext_vector_type(8))) float v8f;

#define HWN 65536
#define WD  256
#define CIN 480

// ---------------------------------------------------------------------------
// Kernel 1: center channel of the 1x1 conv via chained V_WMMA_F32_16X16X4_F32.
// Each wave computes 16 pixels. A(16x4) = broadcast of w17[4k..4k+3] in every
// row; B(4x16) = features for 4 channels x 16 pixels. D row0 lanes 0..15 hold
// the per-pixel dot products. Feature bytes are read exactly once (coalesced).
// ---------------------------------------------------------------------------
__global__ __launch_bounds__(256) void center_wmma(const float* __restrict__ feat,
                                                   const float* __restrict__ cw,
                                                   const float* __restrict__ cb,
                                                   float* __restrict__ cmap) {
    __shared__ float wsm[CIN];
    int tid = threadIdx.x;
    for (int i = tid; i < CIN; i += 256) wsm[i] = cw[17 * CIN + i];
    __syncthreads();

    int lane = tid & 31;
    int wv   = tid >> 5;
    int n    = lane & 15;        // pixel within group
    int kh   = lane >> 4;        // 0: K rows 0,1   1: K rows 2,3
    long p0  = ((long)blockIdx.x * 8 + wv) * 16;
    const float* fb = feat + p0 + n;

    v8f acc = {};
    for (int kb = 0; kb < CIN / 4; ++kb) {
        int c0 = 4 * kb + 2 * kh;
        v2f a, b;
        a.x = wsm[c0];
        a.y = wsm[c0 + 1];
        b.x = fb[(long)c0 * HWN];
        b.y = fb[(long)(c0 + 1) * HWN];
        if (kb + 2 < CIN / 4)                     // pull K-block kb+2 toward L2/L0
            __builtin_prefetch(fb + (long)(c0 + 8) * HWN, 0, 1);
        acc = __builtin_amdgcn_wmma_f32_16x16x4_f32(
            /*neg_a=*/false, a, /*neg_b=*/false, b,
            /*c_mod=*/(short)0, acc, /*reuse_a=*/false, /*reuse_b=*/false);
    }

    float z = acc[0] + cb[17];                    // D[M=0][N=lane] on lanes 0..15
    float s = 1.0f / (1.0f + __expf(-z));
    s = fminf(fmaxf(s, 1e-4f), 1.0f - 1e-4f);
    if (kh == 0) cmap[p0 + n] = s;
}

// ---------------------------------------------------------------------------
// Kernel 2: center = 0.5*(center + avgpool3x3(center))  (zero-padded sum / 9)
// ---------------------------------------------------------------------------
__global__ __launch_bounds__(256) void pool_avg(const float* __restrict__ cmap,
                                                float* __restrict__ c2) {
    int p = blockIdx.x * blockDim.x + threadIdx.x;
    int x = p & (WD - 1), y = p >> 8;
    float sum = 0.0f;
    for (int dy = -1; dy <= 1; ++dy)
        for (int dx = -1; dx <= 1; ++dx) {
            int yy = y + dy, xx = x + dx;
            if (yy >= 0 && yy < WD && xx >= 0 && xx < WD)
                sum += cmap[yy * WD + xx];
        }
    c2[p] = 0.5f * (cmap[p] + sum * (1.0f / 9.0f));
}

// ---------------------------------------------------------------------------
// Kernel 3: NMS — keep value only where it equals its 5x5 max.
// ---------------------------------------------------------------------------
__global__ __launch_bounds__(256) void pool_max_nms(const float* __restrict__ c2,
                                                    float* __restrict__ supp) {
    int p = blockIdx.x * blockDim.x + threadIdx.x;
    int x = p & (WD - 1), y = p >> 8;
    float m = -__builtin_inff();
    for (int dy = -2; dy <= 2; ++dy)
        for (int dx = -2; dx <= 2; ++dx) {
            int yy = y + dy, xx = x + dx;
            if (yy >= 0 && yy < WD && xx >= 0 && xx < WD)
                m = fmaxf(m, c2[yy * WD + xx]);
        }
    float v = c2[p];
    supp[p] = (m == v) ? v : 0.0f;
}

// ---------------------------------------------------------------------------
// Kernel 4: iterative top-30 (single workgroup). Tie-break: lower index wins,
// matching lax.top_k ordering. Extraction order gives descending scores.
// ---------------------------------------------------------------------------
__global__ __launch_bounds__(1024) void topk30(float* __restrict__ supp,
                                               float* __restrict__ outs,
                                               int* __restrict__ outi) {
    __shared__ float sv[1024];
    __shared__ int   si[1024];
    int t = threadIdx.x;
    for (int iter = 0; iter < 30; ++iter) {
        float best = -__builtin_inff();
        int   bi   = 0x7fffffff;
        for (int i = t; i < HWN; i += 1024) {
            float v = supp[i];
            if (v > best || (v == best && i < bi)) { best = v; bi = i; }
        }
        sv[t] = best; si[t] = bi;
        __syncthreads();
        for (int off = 512; off > 0; off >>= 1) {
            if (t < off) {
                float v2 = sv[t + off]; int i2 = si[t + off];
                if (v2 > sv[t] || (v2 == sv[t] && i2 < si[t])) { sv[t] = v2; si[t] = i2; }
            }
            __syncthreads();
        }
        if (t == 0) {
            outs[iter] = sv[0];
            outi[iter] = si[0];
            supp[si[0]] = -__builtin_inff();     // remove for next round
        }
        __syncthreads();
    }
}

// ---------------------------------------------------------------------------
// Kernel 5: write flattened output tuple:
//   [0,60)      instance_coord (y,x) as float
//   [60,90)     scores
//   [90,14490)  instance_param (30 x 480) gathered from features
//   [14490,14520) valid = score > 0.01
// ---------------------------------------------------------------------------
__global__ __launch_bounds__(256) void gather_out(const float* __restrict__ feat,
                                                  const float* __restrict__ outs,
                                                  const int* __restrict__ outi,
                                                  float* __restrict__ out) {
    int i   = blockIdx.x;
    int idx = outi[i];
    float sc = outs[i];
    if (threadIdx.x == 0) {
        out[2 * i]     = (float)(idx >> 8);        // y
        out[2 * i + 1] = (float)(idx & (WD - 1));  // x
        out[60 + i]    = sc;
        out[14490 + i] = (sc > 0.01f) ? 1.0f : 0.0f;
    }
    for (int c = threadIdx.x; c < CIN; c += blockDim.x)
        out[90 + i * CIN + c] = feat[(long)c * HWN + idx];
}

extern "C" void kernel_launch(void* const* d_in, const int* in_sizes, int n_in,
                              void* d_out, int out_size, void* d_ws, size_t ws_size,
                              hipStream_t stream) {
    const float* feat = (const float*)d_in[0];   // (1,480,256,256)
    const float* cw   = (const float*)d_in[1];   // (18,480)
    const float* cb   = (const float*)d_in[2];   // (18,)

    float* ws   = (float*)d_ws;
    float* cmap = ws;                 // 65536
    float* c2   = ws + HWN;           // 65536
    float* supp = ws + 2 * HWN;       // 65536 (mutated by topk, rebuilt each call)
    float* outs = ws + 3 * HWN;       // 30 scores
    int*   outi = (int*)(ws + 3 * HWN + 32);  // 30 indices
    float* out  = (float*)d_out;

    center_wmma<<<512, 256, 0, stream>>>(feat, cw, cb, cmap);
    pool_avg<<<HWN / 256, 256, 0, stream>>>(cmap, c2);
    pool_max_nms<<<HWN / 256, 256, 0, stream>>>(c2, supp);
    topk30<<<1, 1024, 0, stream>>>(supp, outs, outi);
    gather_out<<<30, 256, 0, stream>>>(feat, outs, outi, out);
}